// RepeatEncoder_19542101197436
// MI455X (gfx1250) — compile-verified
//
#include <hip/hip_runtime.h>

typedef float v4f __attribute__((ext_vector_type(4)));

#ifndef __has_builtin
#define __has_builtin(x) 0
#endif

#if __has_builtin(__builtin_amdgcn_global_load_async_to_lds_b32) && \
    __has_builtin(__builtin_amdgcn_s_wait_asynccnt)
#define LIF_USE_ASYNC 1
#else
#define LIF_USE_ASYNC 0
#endif

namespace {
constexpr int kB = 64;
constexpr int kL = 2048;
constexpr int kC = 64;
constexpr int kT = 8;
constexpr int kTileL = 64;   // l-extent of one transpose tile
constexpr int kPad = 65;     // LDS pitch in floats (conflict-free, stride 65 % 64 == 1)
}  // namespace

using gint_t = __attribute__((address_space(1))) int*;   // global dword pointer
using lint_t = __attribute__((address_space(3))) int*;   // LDS dword pointer

// One workgroup handles a 64(l) x 64(c) tile of batch b:
//   phase 1: input tile (C-contiguous) -> LDS, transposed to [c][l]
//            via gfx1250 async load-to-LDS (per-lane LDS dest address)
//   phase 2: per element run the 8-step LIF recurrence, write 8 output planes
//            with L-contiguous 128-bit non-temporal stores.
__global__ __launch_bounds__(256) void lif_repeat_kernel(
    const float* __restrict__ in, float* __restrict__ out) {
  __shared__ float tile[kC * kPad];

  const int tid = threadIdx.x;
  const int b   = blockIdx.y;
  const int l0  = blockIdx.x * kTileL;

  // Tile is 64 rows (l) x 64 cols (c) == 4096 contiguous floats (C == 64).
  const float* src = in + ((size_t)b * kL + l0) * kC;

#if LIF_USE_ASYNC
  {
    gint_t g = (gint_t)src;
    lint_t l = (lint_t)&tile[0];
#pragma unroll
    for (int k = 0; k < 16; ++k) {
      const int idx = tid + 256 * k;     // 0..4095, contiguous in global memory
      const int c   = idx & (kC - 1);
      const int r   = idx >> 6;          // l within tile
      // Hardware scatters into LDS at a per-lane address -> transpose for free.
      __builtin_amdgcn_global_load_async_to_lds_b32(
          g + idx, l + (c * kPad + r), 0, 0);
    }
    __builtin_amdgcn_s_wait_asynccnt(0);
  }
#else
#pragma unroll
  for (int k = 0; k < 4; ++k) {
    const int idx = tid + 256 * k;       // 0..1023 float4 slots
    const int r   = idx >> 4;            // l within tile
    const int c4  = (idx & 15) << 2;     // c base
    const v4f v = __builtin_nontemporal_load(
        reinterpret_cast<const v4f*>(src + r * kC + c4));
    tile[(c4 + 0) * kPad + r] = v.x;
    tile[(c4 + 1) * kPad + r] = v.y;
    tile[(c4 + 2) * kPad + r] = v.z;
    tile[(c4 + 3) * kPad + r] = v.w;
  }
#endif
  __syncthreads();

  constexpr size_t kPlane = (size_t)kB * kC * kL;  // elements per t-plane

#pragma unroll
  for (int k = 0; k < 4; ++k) {
    const int idx = tid + 256 * k;
    const int cc  = idx >> 4;            // output row (c)
    const int l4  = (idx & 15) << 2;     // 4 consecutive l
    const float x0 = tile[cc * kPad + l4 + 0];
    const float x1 = tile[cc * kPad + l4 + 1];
    const float x2 = tile[cc * kPad + l4 + 2];
    const float x3 = tile[cc * kPad + l4 + 3];

    float v0 = 0.f, v1 = 0.f, v2 = 0.f, v3 = 0.f;
    float* dst = out + ((size_t)b * kC + cc) * kL + (l0 + l4);

#pragma unroll
    for (int t = 0; t < kT; ++t) {
      // LIF charge, v_reset = 0, tau = 2  (match reference expression shape)
      v0 = v0 + (x0 - v0) * 0.5f;
      v1 = v1 + (x1 - v1) * 0.5f;
      v2 = v2 + (x2 - v2) * 0.5f;
      v3 = v3 + (x3 - v3) * 0.5f;
      const bool f0 = v0 >= 1.0f;
      const bool f1 = v1 >= 1.0f;
      const bool f2 = v2 >= 1.0f;
      const bool f3 = v3 >= 1.0f;
      v4f s;
      s.x = f0 ? 1.0f : 0.0f;
      s.y = f1 ? 1.0f : 0.0f;
      s.z = f2 ? 1.0f : 0.0f;
      s.w = f3 ? 1.0f : 0.0f;
      v0 = f0 ? 0.0f : v0;   // hard reset
      v1 = f1 ? 0.0f : v1;
      v2 = f2 ? 0.0f : v2;
      v3 = f3 ? 0.0f : v3;
      // Streaming store: output plane is written once, never re-read.
      __builtin_nontemporal_store(
          s, reinterpret_cast<v4f*>(dst + (size_t)t * kPlane));
    }
  }
}

extern "C" void kernel_launch(void* const* d_in, const int* in_sizes, int n_in,
                              void* d_out, int out_size, void* d_ws, size_t ws_size,
                              hipStream_t stream) {
  (void)in_sizes; (void)n_in; (void)out_size; (void)d_ws; (void)ws_size;
  const float* in = (const float*)d_in[0];
  float* out = (float*)d_out;
  dim3 grid(kL / kTileL, kB);   // (32, 64) = 2048 workgroups
  lif_repeat_kernel<<<grid, dim3(256), 0, stream>>>(in, out);
}